// Network_14096082666295
// MI455X (gfx1250) — compile-verified
//
#include <hip/hip_runtime.h>

// Reference reduces to a pure gather/broadcast (zero FLOPs):
//   out[0, t, j, 0:10] = [x[0,0,chA,t], x[0,1,chA,t], a, chA,
//                         x[0,0,chB,t], x[0,1,chB,t], b, chB, rc0, rc1]
// (a,b) = gi_seg[i], chA/chB from the segment's chans tables, seg = j/12000.
// Output (1,512,36000,10) fp32 = 737 MB -> store-bandwidth-bound,
// floor ~32us @ 23.3 TB/s. x (6.3 MB) stays L2-resident (RT loads);
// output streamed with non-temporal stores so it never evicts x.
//
// This version processes an even/odd row PAIR per thread: 80 B per t,
// 16B-aligned -> 5x global_store_b128 (NT) per t instead of 10x b64,
// and gi/rc metadata loads become single int4/float4 loads.

typedef float v2f __attribute__((ext_vector_type(2)));
typedef float v4f __attribute__((ext_vector_type(4)));
typedef int   v4i __attribute__((ext_vector_type(4)));

#define T_DIM   512
#define NCH     1536
#define FSTRIDE (NCH * T_DIM)    // stride between f=0 and f=1 planes of x
#define NSEG    12000
#define NJ      (3 * NSEG)
#define NPAIR   (NJ / 2)         // 18000 row pairs
#define T_TILE  64
#define ROWF    10               // floats per output row

__global__ __launch_bounds__(256)
void crossings_gather_kernel(const float* __restrict__ x,
                             const int*   __restrict__ chans_p0,
                             const int*   __restrict__ chans_p1,
                             const int*   __restrict__ chans_p2,
                             const int*   __restrict__ gi_01,
                             const int*   __restrict__ gi_12,
                             const int*   __restrict__ gi_20,
                             const float* __restrict__ rc_01,
                             const float* __restrict__ rc_12,
                             const float* __restrict__ rc_20,
                             float*       __restrict__ out)
{
    const int jp = blockIdx.x * blockDim.x + threadIdx.x;  // row-pair id
    if (jp >= NPAIR) return;
    const int r0    = 2 * jp;                              // even row; r1 = r0+1
    const int tbase = blockIdx.y * T_TILE;

    // ---- per-pair metadata (t-invariant; segment size 12000 is even, so an
    //      even/odd pair never straddles a segment boundary) ----
    const int*   gi;  const float* rc;
    const int*   chA_tab; const int* chB_tab;
    int i0;
    if (r0 < NSEG)          { i0 = r0;            gi = gi_01; rc = rc_01; chA_tab = chans_p0; chB_tab = chans_p1; }
    else if (r0 < 2 * NSEG) { i0 = r0 - NSEG;     gi = gi_12; rc = rc_12; chA_tab = chans_p1; chB_tab = chans_p2; }
    else                    { i0 = r0 - 2 * NSEG; gi = gi_20; rc = rc_20; chA_tab = chans_p2; chB_tab = chans_p0; }

    // i0 even -> 2*i0 multiple of 4 -> 16B-aligned vector loads
    const v4i g = *(const v4i*)(gi + 2 * i0);   // {a0, b0, a1, b1}
    const v4f r = *(const v4f*)(rc + 2 * i0);   // {rc00, rc01, rc10, rc11}

    const int a0 = g.x, b0 = g.y, a1 = g.z, b1 = g.w;
    const int chA0 = chA_tab[a0], chB0 = chB_tab[b0];
    const int chA1 = chA_tab[a1], chB1 = chB_tab[b1];

    const float fA0w = (float)a0, fA0c = (float)chA0;
    const float fB0w = (float)b0, fB0c = (float)chB0;
    const float fA1w = (float)a1, fA1c = (float)chA1;
    const float fB1w = (float)b1, fB1c = (float)chB1;

    // ---- eight sequential x streams for this pair (L2-resident) ----
    const float* pa00 = x + (size_t)chA0 * T_DIM + tbase;  // f=0, chA0
    const float* pa01 = pa00 + FSTRIDE;                    // f=1, chA0
    const float* pb00 = x + (size_t)chB0 * T_DIM + tbase;
    const float* pb01 = pb00 + FSTRIDE;
    const float* pa10 = x + (size_t)chA1 * T_DIM + tbase;
    const float* pa11 = pa10 + FSTRIDE;
    const float* pb10 = x + (size_t)chB1 * T_DIM + tbase;
    const float* pb11 = pb10 + FSTRIDE;

    // base of the 80B pair-row; (t*36000 + 2*jp)*40 bytes is 16B-aligned
    float* const obase = out + ((size_t)tbase * NJ + (size_t)r0) * ROWF;
    const size_t ostride = (size_t)NJ * ROWF;              // floats per t step

    for (int tt = 0; tt < T_TILE; tt += 4) {
        // gfx1250 prefetch of the next 16-float chunk of each stream
        __builtin_prefetch(pa00 + tt + 16, 0, 3);
        __builtin_prefetch(pa01 + tt + 16, 0, 3);
        __builtin_prefetch(pb00 + tt + 16, 0, 3);
        __builtin_prefetch(pb01 + tt + 16, 0, 3);
        __builtin_prefetch(pa10 + tt + 16, 0, 3);
        __builtin_prefetch(pa11 + tt + 16, 0, 3);
        __builtin_prefetch(pb10 + tt + 16, 0, 3);
        __builtin_prefetch(pb11 + tt + 16, 0, 3);

        const v4f va00 = *(const v4f*)(pa00 + tt);  // 16B-aligned (t % 4 == 0)
        const v4f va01 = *(const v4f*)(pa01 + tt);
        const v4f vb00 = *(const v4f*)(pb00 + tt);
        const v4f vb01 = *(const v4f*)(pb01 + tt);
        const v4f va10 = *(const v4f*)(pa10 + tt);
        const v4f va11 = *(const v4f*)(pa11 + tt);
        const v4f vb10 = *(const v4f*)(pb10 + tt);
        const v4f vb11 = *(const v4f*)(pb11 + tt);

#pragma unroll
        for (int k = 0; k < 4; ++k) {
            float* o = obase + (size_t)(tt + k) * ostride;  // 16B-aligned
            const v4f s0 = { va00[k], va01[k], fA0w, fA0c };          // row0: xA, idsA
            const v4f s1 = { vb00[k], vb01[k], fB0w, fB0c };          // row0: xB, idsB
            const v4f s2 = { r.x,     r.y,     va10[k], va11[k] };    // row0 rc | row1 xA
            const v4f s3 = { fA1w,    fA1c,    vb10[k], vb11[k] };    // row1 idsA, xB
            const v4f s4 = { fB1w,    fB1c,    r.z,     r.w };        // row1 idsB, rc
            // Non-temporal B128 stores: output is write-once streaming
            __builtin_nontemporal_store(s0, (v4f*)(o + 0));
            __builtin_nontemporal_store(s1, (v4f*)(o + 4));
            __builtin_nontemporal_store(s2, (v4f*)(o + 8));
            __builtin_nontemporal_store(s3, (v4f*)(o + 12));
            __builtin_nontemporal_store(s4, (v4f*)(o + 16));
        }
    }
}

extern "C" void kernel_launch(void* const* d_in, const int* in_sizes, int n_in,
                              void* d_out, int out_size, void* d_ws, size_t ws_size,
                              hipStream_t stream) {
    (void)in_sizes; (void)n_in; (void)out_size; (void)d_ws; (void)ws_size;

    // setup_inputs() order:
    // 0:x 1:wires_p0 2:chans_p0 3:wires_p1 4:chans_p1 5:wires_p2 6:chans_p2
    // 7:gi_01 8:gi_12 9:gi_20 10:rc_01 11:rc_12 12:rc_20
    const float* x        = (const float*)d_in[0];
    const int*   chans_p0 = (const int*)  d_in[2];
    const int*   chans_p1 = (const int*)  d_in[4];
    const int*   chans_p2 = (const int*)  d_in[6];
    const int*   gi_01    = (const int*)  d_in[7];
    const int*   gi_12    = (const int*)  d_in[8];
    const int*   gi_20    = (const int*)  d_in[9];
    const float* rc_01    = (const float*)d_in[10];
    const float* rc_12    = (const float*)d_in[11];
    const float* rc_20    = (const float*)d_in[12];
    float*       out      = (float*)d_out;

    dim3 block(256, 1, 1);
    dim3 grid((NPAIR + 255) / 256, T_DIM / T_TILE, 1);   // 71 x 8 blocks
    crossings_gather_kernel<<<grid, block, 0, stream>>>(
        x, chans_p0, chans_p1, chans_p2,
        gi_01, gi_12, gi_20, rc_01, rc_12, rc_20, out);
}